// E2GN2Layer_72138270704317
// MI455X (gfx1250) — compile-verified
//
#include <hip/hip_runtime.h>
#include <hip/hip_bf16.h>
#include <stddef.h>

typedef __attribute__((ext_vector_type(2))) float v2f;
typedef __attribute__((ext_vector_type(8))) float v8f;

#define B_ 4096
#define N_ 256
#define D_ 64
#define C_ 3
#define M_ 32

static __device__ __forceinline__ float silu_f(float x) {
    return x / (1.0f + __expf(-x));
}

// D = A(16x4 f32) * B(4x16 f32) + C(16x16 f32), wave32, one matrix per wave.
static __device__ __forceinline__ v8f wmma16x16x4(v2f a, v2f b, v8f c) {
    return __builtin_amdgcn_wmma_f32_16x16x4_f32(
        /*neg_a=*/false, a, /*neg_b=*/false, b,
        /*c_mod=*/(short)0, c, /*reuse_a=*/false, /*reuse_b=*/false);
}

__global__ __launch_bounds__(256)
void egnn_layer_kernel(const float* __restrict__ hi, const float* __restrict__ xi,
                       const float* __restrict__ hj, const float* __restrict__ xj,
                       const float* __restrict__ We, const float* __restrict__ be,
                       const float* __restrict__ Wx, const float* __restrict__ bx,
                       const float* __restrict__ Wx2, const float* __restrict__ bx2,
                       const float* __restrict__ Wh, const float* __restrict__ bh,
                       float* __restrict__ out_hi, float* __restrict__ out_xi,
                       float* __restrict__ out_hj, float* __restrict__ out_xj)
{
    __shared__ float s_ud[N_][C_];     // u_diff
    __shared__ float s_u2[N_][C_];     // u_diff^2
    __shared__ float s_base[M_];       // hi @ We_hi + be
    __shared__ float s_mi[M_];
    __shared__ float s_hi[D_];
    __shared__ float s_mip[32][M_];    // deterministic mi partials
    __shared__ float s_xp[32][C_];     // deterministic x partials
    __shared__ float s_phix2;

    const int b    = blockIdx.x;
    const int tid  = threadIdx.x;
    const int wave = tid >> 5;
    const int lane = tid & 31;
    const int half = lane >> 4;   // 0: K=0,1 | 1: K=2,3 (f32 WMMA A/B layout)
    const int l16  = lane & 15;

    // ---- Phase 0: xj copy (fused, non-temporal stream), u_diff/u2, base ----
    if (tid < D_) s_hi[tid] = hi[(size_t)b * D_ + tid];

    for (int i = 0; i < 3; ++i) {
        int idx = tid + i * 256;                       // 0..767 = N*C
        size_t g = (size_t)b * (N_ * C_) + idx;
        float v = __builtin_nontemporal_load(xj + g);
        __builtin_nontemporal_store(v, out_xj + g);    // pass-through output
        int n = idx / 3, c = idx - n * 3;
        float u = xi[(size_t)b * C_ + c] - v;
        s_ud[n][c] = u;
        s_u2[n][c] = u * u;
    }
    if (tid < M_) {
        float acc = be[tid];
        #pragma unroll 8
        for (int d = 0; d < D_; ++d)
            acc += hi[(size_t)b * D_ + d] * We[d * M_ + tid];
        s_base[tid] = acc;
    }
    __syncthreads();

    // ---- Phase 1: preload B-fragments (We_hj rows 64..127, We_u rows 128..130) ----
    const int m0 = l16;
    const int m1 = 16 + l16;
    v2f bf0[16], bf1[16];
    #pragma unroll
    for (int k = 0; k < 16; ++k) {
        int row = D_ + 4 * k + 2 * half;
        v2f t0; t0.x = We[row * M_ + m0]; t0.y = We[(row + 1) * M_ + m0];
        v2f t1; t1.x = We[row * M_ + m1]; t1.y = We[(row + 1) * M_ + m1];
        bf0[k] = t0; bf1[k] = t1;
    }
    v2f bu0, bu1;
    if (half == 0) {
        bu0.x = We[128 * M_ + m0]; bu0.y = We[129 * M_ + m0];
        bu1.x = We[128 * M_ + m1]; bu1.y = We[129 * M_ + m1];
    } else {
        bu0.x = We[130 * M_ + m0]; bu0.y = 0.0f;
        bu1.x = We[130 * M_ + m1]; bu1.y = 0.0f;
    }
    const float wx0 = Wx[m0], wx1 = Wx[m1];
    const float bxv = bx[0];

    // ---- Phase 2: WMMA over 16-row tiles; fused NT hj copy; silu; reductions ----
    #pragma unroll
    for (int pass = 0; pass < 2; ++pass) {
        const int blk = pass * 8 + wave;               // 0..15
        const int n0  = blk * 16;
        v8f acc0 = {0.f,0.f,0.f,0.f,0.f,0.f,0.f,0.f};
        v8f acc1 = {0.f,0.f,0.f,0.f,0.f,0.f,0.f,0.f};

        const size_t rowbase = ((size_t)b * N_ + (n0 + l16)) * D_ + 2 * half;
        #pragma unroll
        for (int k = 0; k < 16; ++k) {
            // A fragment (b64, non-temporal: touch-once 256 MB stream)
            v2f a = __builtin_nontemporal_load((const v2f*)(hj + rowbase + 4 * k));
            acc0 = wmma16x16x4(a, bf0[k], acc0);
            acc1 = wmma16x16x4(a, bf1[k], acc1);
            // fused pass-through copy from the same registers
            __builtin_nontemporal_store(a, (v2f*)(out_hj + rowbase + 4 * k));
        }
        // u_2norm @ We_u folded in as one extra K=4 step (4th element zero-padded)
        v2f au;
        if (half == 0) { au.x = s_u2[n0 + l16][0]; au.y = s_u2[n0 + l16][1]; }
        else           { au.x = s_u2[n0 + l16][2]; au.y = 0.0f; }
        acc0 = wmma16x16x4(au, bu0, acc0);
        acc1 = wmma16x16x4(au, bu1, acc1);

        // mij = silu(acc + base); row n = n0 + r + 8*half, col = m0 / m1
        const float bb0 = s_base[m0], bb1 = s_base[m1];
        float sum0 = 0.f, sum1 = 0.f;
        float xl0 = 0.f, xl1 = 0.f, xl2 = 0.f;
        #pragma unroll
        for (int r = 0; r < 8; ++r) {
            float v0 = silu_f(acc0[r] + bb0);
            float v1 = silu_f(acc1[r] + bb1);
            sum0 += v0; sum1 += v1;
            // phi_x row dot: reduce across the 16-lane half (m = 0..31)
            float t = v0 * wx0 + v1 * wx1;
            t += __shfl_xor(t, 1, 32);
            t += __shfl_xor(t, 2, 32);
            t += __shfl_xor(t, 4, 32);
            t += __shfl_xor(t, 8, 32);
            float phi = silu_f(t + bxv);
            int n = n0 + r + 8 * half;
            xl0 += s_ud[n][0] * phi;
            xl1 += s_ud[n][1] * phi;
            xl2 += s_ud[n][2] * phi;
        }
        const int slot = blk * 2 + half;               // 0..31
        s_mip[slot][m0] = sum0;
        s_mip[slot][m1] = sum1;
        if (l16 == 0) { s_xp[slot][0] = xl0; s_xp[slot][1] = xl1; s_xp[slot][2] = xl2; }
    }
    __syncthreads();

    // ---- Phase 3: deterministic final reductions + small projections ----
    if (tid < M_) {
        float mi = 0.f;
        #pragma unroll
        for (int i = 0; i < 32; ++i) mi += s_mip[i][tid];
        s_mi[tid] = mi;
        float p = mi * Wx2[tid];
        p += __shfl_xor(p, 1, 32);
        p += __shfl_xor(p, 2, 32);
        p += __shfl_xor(p, 4, 32);
        p += __shfl_xor(p, 8, 32);
        p += __shfl_xor(p, 16, 32);
        if (tid == 0) s_phix2 = silu_f(p + bx2[0]);
    }
    __syncthreads();

    if (tid < C_) {
        float xs = 0.f;
        #pragma unroll
        for (int i = 0; i < 32; ++i) xs += s_xp[i][tid];
        out_xi[(size_t)b * C_ + tid] =
            xi[(size_t)b * C_ + tid] * s_phix2 + xs * (1.0f / (float)N_);
    }
    if (tid < D_) {
        float acc = bh[tid];
        #pragma unroll 8
        for (int k = 0; k < D_; ++k) acc += s_hi[k] * Wh[k * D_ + tid];
        #pragma unroll 8
        for (int k = 0; k < M_; ++k) acc += s_mi[k] * Wh[(D_ + k) * D_ + tid];
        out_hi[(size_t)b * D_ + tid] = silu_f(acc);
    }
}

extern "C" void kernel_launch(void* const* d_in, const int* in_sizes, int n_in,
                              void* d_out, int out_size, void* d_ws, size_t ws_size,
                              hipStream_t stream) {
    const float* hi  = (const float*)d_in[0];
    const float* xi  = (const float*)d_in[1];
    const float* hj  = (const float*)d_in[2];
    const float* xj  = (const float*)d_in[3];
    const float* We  = (const float*)d_in[4];
    const float* be  = (const float*)d_in[5];
    const float* Wx  = (const float*)d_in[6];
    const float* bx  = (const float*)d_in[7];
    const float* Wx2 = (const float*)d_in[8];
    const float* bx2 = (const float*)d_in[9];
    const float* Wh  = (const float*)d_in[10];
    const float* bh  = (const float*)d_in[11];

    float* out    = (float*)d_out;
    float* out_hi = out;                                   // (B, D)
    float* out_xi = out_hi + (size_t)B_ * D_;              // (B, C)
    float* out_hj = out_xi + (size_t)B_ * C_;              // (B, N, D)
    float* out_xj = out_hj + (size_t)B_ * N_ * D_;         // (B, N, C)

    egnn_layer_kernel<<<B_, 256, 0, stream>>>(
        hi, xi, hj, xj, We, be, Wx, bx, Wx2, bx2, Wh, bh,
        out_hi, out_xi, out_hj, out_xj);
}